// ElasticEmbedding_61555471286588
// MI455X (gfx1250) — compile-verified
//
#include <hip/hip_runtime.h>
#include <stdint.h>

typedef __attribute__((ext_vector_type(4))) float v4f;
typedef __attribute__((ext_vector_type(4))) int   v4i;

#define EMBED        768
#define V4_PER_ROW   192            // 768 floats / 4 per v4f
#define WAVES_PER_BLOCK 8           // 256 threads, wave32

#if defined(__has_builtin)
#if __has_builtin(__builtin_amdgcn_global_load_async_to_lds_b128) && \
    __has_builtin(__builtin_amdgcn_s_wait_asynccnt)
#define HAVE_ASYNC 1
#endif
#endif

// ---------------------------------------------------------------------------
// Rightmost-occurrence binary search over the *sorted* residual_index.
// Returns residual slot (last duplicate wins) or -1.
// ---------------------------------------------------------------------------
__device__ __forceinline__ int rightmost_slot(const int* __restrict__ ridx,
                                              int R, int xi) {
  int lo = 0, hi = R;                    // upper_bound
  while (lo < hi) {
    int mid = (lo + hi) >> 1;
    if (ridx[mid] <= xi) lo = mid + 1; else hi = mid;
  }
  return (lo > 0 && ridx[lo - 1] == xi) ? (lo - 1) : -1;
}

// ---------------------------------------------------------------------------
// Phase 1: resolve each token's source row pointer once (removes the
// dependent x -> slot -> row latency chain from the bandwidth kernel).
// ---------------------------------------------------------------------------
__global__ __launch_bounds__(256) void k_build_ptr(
    const int* __restrict__ x,
    const float* __restrict__ pre,
    const float* __restrict__ res,
    const int* __restrict__ ridx,
    int R, int ntok,
    const float** __restrict__ srcptr) {
  int t = blockIdx.x * blockDim.x + threadIdx.x;
  if (t >= ntok) return;
  int xi = x[t];
  int r  = rightmost_slot(ridx, R, xi);
  srcptr[t] = (r >= 0) ? (res + (size_t)r * EMBED)
                       : (pre + (size_t)xi * EMBED);
}

// ---------------------------------------------------------------------------
// Phase 2a (CDNA5 async path): one wave copies one 3 KB row per iteration.
// Double-buffered GLOBAL_LOAD_ASYNC_TO_LDS_B128 pipeline, drained with
// s_wait_asynccnt; output written with non-temporal 128-bit stores so the
// 96 MB streamed output does not evict embedding rows from the 192 MB L2.
// ---------------------------------------------------------------------------
#if HAVE_ASYNC
typedef __attribute__((address_space(1))) v4i gv4i;   // printed as "__device__"
typedef __attribute__((address_space(3))) v4i sv4i;   // printed as "__shared__"

__device__ __forceinline__ gv4i* to_global(const void* p) {
  // AS1 pointers share the 64-bit value of generic global pointers.
  return (gv4i*)(uintptr_t)p;
}
__device__ __forceinline__ sv4i* to_lds(void* p) {
  // Generic pointer to __shared__ = {shared-aperture hi32, LDS offset lo32};
  // truncating to 32 bits recovers the addrspace(3) value (ISA aperture rules).
  return (sv4i*)(uint32_t)(uintptr_t)p;
}

__device__ __forceinline__ void async_row_copy(const float* __restrict__ gsrc,
                                               v4f* lbuf, int lane) {
  const char* g = (const char*)gsrc + lane * 16;
  char*       l = (char*)lbuf       + lane * 16;
  // 6 x (32 lanes x 16 B) = 3072 B = one embedding row
  __builtin_amdgcn_global_load_async_to_lds_b128(to_global(g +    0), to_lds(l +    0), 0, 0);
  __builtin_amdgcn_global_load_async_to_lds_b128(to_global(g +  512), to_lds(l +  512), 0, 0);
  __builtin_amdgcn_global_load_async_to_lds_b128(to_global(g + 1024), to_lds(l + 1024), 0, 0);
  __builtin_amdgcn_global_load_async_to_lds_b128(to_global(g + 1536), to_lds(l + 1536), 0, 0);
  __builtin_amdgcn_global_load_async_to_lds_b128(to_global(g + 2048), to_lds(l + 2048), 0, 0);
  __builtin_amdgcn_global_load_async_to_lds_b128(to_global(g + 2560), to_lds(l + 2560), 0, 0);
}

__global__ __launch_bounds__(256) void k_gather_async(
    const float* const* __restrict__ srcptr,
    float* __restrict__ out, int ntok) {
  __shared__ v4f lds[2][WAVES_PER_BLOCK][V4_PER_ROW];   // 48 KB static LDS
  const int lane   = threadIdx.x & 31;
  const int wave   = threadIdx.x >> 5;
  const int stride = gridDim.x * WAVES_PER_BLOCK;
  int t   = blockIdx.x * WAVES_PER_BLOCK + wave;
  int buf = 0;

  if (t < ntok) async_row_copy(srcptr[t], &lds[0][wave][0], lane);

  for (; t < ntok; t += stride) {
    int tn = t + stride;
    if (tn < ntok) {
      async_row_copy(srcptr[tn], &lds[buf ^ 1][wave][0], lane);
      __builtin_amdgcn_s_wait_asynccnt(6);   // async loads retire in order:
                                             // the 6 for token t are done
    } else {
      __builtin_amdgcn_s_wait_asynccnt(0);
    }
    asm volatile("" ::: "memory");           // keep LDS reads below the wait

    float*     o = out + (size_t)t * EMBED;
    const v4f* l = &lds[buf][wave][0];
#pragma unroll
    for (int k = 0; k < 6; ++k) {
      v4f v = l[lane + 32 * k];
      __builtin_nontemporal_store(v, (v4f*)o + lane + 32 * k);
    }
    buf ^= 1;
  }
}
#endif // HAVE_ASYNC

// ---------------------------------------------------------------------------
// Phase 2b (fallback): flat one-v4f-per-thread gather, NT stores.
// ---------------------------------------------------------------------------
__global__ __launch_bounds__(256) void k_gather_direct(
    const float* const* __restrict__ srcptr,
    float* __restrict__ out, int ntok) {
  size_t idx   = blockIdx.x * (size_t)blockDim.x + threadIdx.x;
  size_t total = (size_t)ntok * V4_PER_ROW;
  if (idx >= total) return;
  int t = (int)(idx / V4_PER_ROW);
  int c = (int)(idx % V4_PER_ROW);
  const v4f* src = (const v4f*)srcptr[t];
  __builtin_nontemporal_store(src[c], (v4f*)out + idx);
}

// Fallback when d_ws is too small for the pointer table: inline search.
__global__ __launch_bounds__(256) void k_gather_flat_search(
    const int* __restrict__ x,
    const float* __restrict__ pre,
    const float* __restrict__ res,
    const int* __restrict__ ridx,
    int R, int ntok,
    float* __restrict__ out) {
  size_t idx   = blockIdx.x * (size_t)blockDim.x + threadIdx.x;
  size_t total = (size_t)ntok * V4_PER_ROW;
  if (idx >= total) return;
  int t = (int)(idx / V4_PER_ROW);
  int c = (int)(idx % V4_PER_ROW);
  int xi = x[t];
  int r  = rightmost_slot(ridx, R, xi);
  const v4f* src = (const v4f*)((r >= 0) ? (res + (size_t)r * EMBED)
                                         : (pre + (size_t)xi * EMBED));
  __builtin_nontemporal_store(src[c], (v4f*)out + idx);
}

// ---------------------------------------------------------------------------
extern "C" void kernel_launch(void* const* d_in, const int* in_sizes, int n_in,
                              void* d_out, int out_size, void* d_ws, size_t ws_size,
                              hipStream_t stream) {
  const int*   x    = (const int*)d_in[0];     // [B*S] token ids
  const float* pre  = (const float*)d_in[1];   // [V, 768]
  const float* res  = (const float*)d_in[2];   // [R, 768]
  const int*   ridx = (const int*)d_in[3];     // [R] sorted
  float*       out  = (float*)d_out;

  const int ntok = in_sizes[0];
  const int R    = in_sizes[3];

  const size_t need = (size_t)ntok * sizeof(float*);
  if (ws_size >= need) {
    const float** srcptr = (const float**)d_ws;
    k_build_ptr<<<(ntok + 255) / 256, 256, 0, stream>>>(x, pre, res, ridx, R,
                                                        ntok, srcptr);
#if HAVE_ASYNC
    int blocks = (ntok + WAVES_PER_BLOCK - 1) / WAVES_PER_BLOCK;
    if (blocks > 1024) blocks = 1024;          // ~4 tokens/wave of pipelining
    k_gather_async<<<blocks, 256, 0, stream>>>(srcptr, out, ntok);
#else
    size_t total = (size_t)ntok * V4_PER_ROW;
    k_gather_direct<<<(total + 255) / 256, 256, 0, stream>>>(srcptr, out, ntok);
#endif
  } else {
    size_t total = (size_t)ntok * V4_PER_ROW;
    k_gather_flat_search<<<(total + 255) / 256, 256, 0, stream>>>(
        x, pre, res, ridx, R, ntok, out);
  }
}